// GCN_9414568312940
// MI455X (gfx1250) — compile-verified
//
#include <hip/hip_runtime.h>
#include <hip/hip_bf16.h>

// ---------------------------------------------------------------------------
// 2-layer GCN for MI455X (gfx1250, wave32).
//  - GEMMs via V_WMMA_F32_16X16X4_F32 (exact fp32; the workload is edge-scatter
//    bandwidth bound, so no precision downgrade is warranted). GEMM is
//    templatized on the output width so all B-loads / C-stores use constant
//    immediate offsets (no 64-bit address VALU in the hot loop), and the
//    K=128 loop is fully unrolled into 32 back-to-back WMMAs.
//  - A-tiles staged into LDS with GLOBAL_LOAD_ASYNC_TO_LDS_B128 (ASYNCcnt).
//  - Edge aggregation: one wave32 per edge, lanes stride channels, per-DWORD
//    global_atomic_add_f32 (no return). H + accumulator fit in 192MB L2.
// ---------------------------------------------------------------------------

typedef __attribute__((ext_vector_type(2))) float v2f;
typedef __attribute__((ext_vector_type(8))) float v8f;

#define KDIM 128          // inner dimension of both GEMMs (IN_CH == HID_CH == 128)
#define LDSP 132          // padded LDS row stride (floats): m*132 % 64 = 4m -> no bank conflicts
#define ASYNC_STAGE 1     // stage A tile via global_load_async_to_lds_b128

#if defined(__has_builtin)
#  if __has_builtin(__builtin_amdgcn_wmma_f32_16x16x4_f32)
#    define HAVE_WMMA_F32X4 1
#  endif
#endif

// ---------------- degree / normalization ----------------

__global__ void gcn_deg_init(float* __restrict__ dinv, int N) {
    int i = blockIdx.x * blockDim.x + threadIdx.x;
    if (i < N) dinv[i] = 1.0f;               // self-loop contributes 1 to degree
}

__global__ void gcn_deg_acc(const int* __restrict__ dst, float* __restrict__ dinv, int E) {
    int e = blockIdx.x * blockDim.x + threadIdx.x;
    if (e < E) atomicAdd(&dinv[dst[e]], 1.0f);
}

__global__ void gcn_deg_fin(float* __restrict__ dinv, int N) {
    int i = blockIdx.x * blockDim.x + threadIdx.x;
    if (i < N) dinv[i] = rsqrtf(dinv[i]);    // deg >= 1 always (self-loop)
}

// ---------------- WMMA GEMM: H[M x NW] = A[M x 128] * W[128 x NW] ----------------
// Block = 128 threads = 4 waves; each block owns a 16-row stripe of A (staged in
// LDS once, asynchronously), waves round-robin the NW/16 column tiles.
// NW is a compile-time constant so strides fold into immediate offsets.

template <int NW>
__global__ void __launch_bounds__(128)
gcn_gemm_wmma(const float* __restrict__ A, const float* __restrict__ W,
              float* __restrict__ H, int M)
{
    constexpr int NTILES = NW / 16;
    __shared__ float lds[16 * LDSP];
    const int row0 = blockIdx.x * 16;

    // Stage the 16 x 128 A tile: 512 float4 slots, 4 per thread.
    for (int i = threadIdx.x; i < 16 * (KDIM / 4); i += 128) {
        const int r  = i >> 5;               // row 0..15
        const int c4 = i & 31;               // float4 index 0..31
        const int rs = (row0 + r < M) ? (row0 + r) : (M - 1);  // clamp (harmless dup)
        const float* gsrc = A + (size_t)rs * KDIM + c4 * 4;
#if ASYNC_STAGE
        // Lane-level 16B DMA straight into LDS, tracked by ASYNCcnt (no VGPR bounce).
        const unsigned int lds_dst = (unsigned int)(uintptr_t)(&lds[r * LDSP + c4 * 4]);
        const unsigned long long gaddr = (unsigned long long)(uintptr_t)gsrc;
        asm volatile("global_load_async_to_lds_b128 %0, %1, off"
                     :: "v"(lds_dst), "v"(gaddr) : "memory");
#else
        const float4 v = *(const float4*)gsrc;
        float* d = &lds[r * LDSP + c4 * 4];
        d[0] = v.x; d[1] = v.y; d[2] = v.z; d[3] = v.w;
#endif
    }
#if ASYNC_STAGE
    asm volatile("s_wait_asynccnt 0" ::: "memory");
#endif
    __syncthreads();

    const int lane = threadIdx.x & 31;
    const int wv   = threadIdx.x >> 5;
    const int m    = lane & 15;              // M index (A) / N index (B,C)
    const int kh   = lane >> 4;              // 0: lanes 0-15, 1: lanes 16-31
    const bool full_tile = (row0 + 16 <= M); // block-uniform -> straight-line stores

#pragma unroll
    for (int nt = wv; nt < NTILES; nt += 4) {
        const float* Wc = W + nt * 16 + m;   // column base; k-strides are immediates
#if HAVE_WMMA_F32X4
        v8f acc = {};
#pragma unroll
        for (int k0 = 0; k0 < KDIM; k0 += 4) {
            const int ka = k0 + 2 * kh;      // A layout: lanes0-15 K=0,1; lanes16-31 K=2,3
            v2f a, b;
            a.x = lds[m * LDSP + ka];
            a.y = lds[m * LDSP + ka + 1];
            b.x = Wc[(k0 + 0) * NW + 2 * kh * NW];          // row ka
            b.y = Wc[(k0 + 1) * NW + 2 * kh * NW];          // row ka+1
            acc = __builtin_amdgcn_wmma_f32_16x16x4_f32(
                      /*neg_a=*/false, a, /*neg_b=*/false, b,
                      /*c_mod=*/(short)0, acc, /*reuse_a=*/false, /*reuse_b=*/false);
        }
#else
        float accs[8] = {};
        for (int k = 0; k < KDIM; ++k) {
            const float bv = Wc[k * NW];
#pragma unroll
            for (int r = 0; r < 8; ++r)
                accs[r] += lds[(kh * 8 + r) * LDSP + k] * bv;
        }
        v8f acc;
#pragma unroll
        for (int r = 0; r < 8; ++r) acc[r] = accs[r];
#endif
        // C layout: VGPR r -> M = r (lanes 0-15) / M = 8+r (lanes 16-31), N = lane%16
        float* Hc = H + (size_t)(row0 + kh * 8) * NW + nt * 16 + m;
        if (full_tile) {
#pragma unroll
            for (int r = 0; r < 8; ++r)
                Hc[r * NW] = acc[r];         // constant immediate offsets
        } else {
#pragma unroll
            for (int r = 0; r < 8; ++r) {
                if (row0 + kh * 8 + r < M) Hc[r * NW] = acc[r];
            }
        }
    }
}

// ---------------- self-loop init (+bias): out[i,:] = H[i,:]*dinv[i]^2 + b ----------------

__global__ void gcn_self_init(const float* __restrict__ H, const float* __restrict__ bias,
                              const float* __restrict__ dinv, float* __restrict__ out,
                              int n4, int c4shift)   // n4 = N*C/4, c4 count = 1<<c4shift
{
    int idx = blockIdx.x * blockDim.x + threadIdx.x;
    if (idx >= n4) return;
    const int node = idx >> c4shift;
    const int c4   = idx & ((1 << c4shift) - 1);
    const float d  = dinv[node];
    const float d2 = d * d;
    const float4 hv = ((const float4*)H)[idx];
    const float4 bv = ((const float4*)bias)[c4];
    float4 o;
    o.x = hv.x * d2 + bv.x;
    o.y = hv.y * d2 + bv.y;
    o.z = hv.z * d2 + bv.z;
    o.w = hv.w * d2 + bv.w;
    ((float4*)out)[idx] = o;
}

// ---------------- edge scatter: out[dst,:] += H[src,:] * dinv[src]*dinv[dst] ----------------
// One wave32 per edge; wave-uniform indices, lanes stride channels (coalesced
// 128B bursts). Prefetch the next edge's source row (global_prefetch_b8).

__global__ void __launch_bounds__(256)
gcn_edge_scatter(const float* __restrict__ H, const int* __restrict__ src,
                 const int* __restrict__ dst, const float* __restrict__ dinv,
                 float* __restrict__ out, int E, int C)
{
    const int lane = threadIdx.x & 31;
    const int gw   = (int)((blockIdx.x * blockDim.x + threadIdx.x) >> 5);
    const int nw   = (int)((gridDim.x * blockDim.x) >> 5);
    for (int e = gw; e < E; e += nw) {
        const int r = src[e];
        const int c = dst[e];
        const int en = e + nw;
        if (en < E)
            __builtin_prefetch(H + (size_t)src[en] * C + lane, 0, 0);
        const float w = dinv[r] * dinv[c];
        const float* hr = H + (size_t)r * C;
        float* oc = out + (size_t)c * C;
        for (int ch = lane; ch < C; ch += 32)
            atomicAdd(&oc[ch], hr[ch] * w);
    }
}

// ---------------- ReLU in place ----------------

__global__ void gcn_relu(float* __restrict__ p, int n4) {
    int idx = blockIdx.x * blockDim.x + threadIdx.x;
    if (idx >= n4) return;
    float4 v = ((float4*)p)[idx];
    v.x = fmaxf(v.x, 0.0f);
    v.y = fmaxf(v.y, 0.0f);
    v.z = fmaxf(v.z, 0.0f);
    v.w = fmaxf(v.w, 0.0f);
    ((float4*)p)[idx] = v;
}

// ---------------- launch ----------------

extern "C" void kernel_launch(void* const* d_in, const int* in_sizes, int n_in,
                              void* d_out, int out_size, void* d_ws, size_t ws_size,
                              hipStream_t stream)
{
    const float* x  = (const float*)d_in[0];
    const float* W1 = (const float*)d_in[1];
    const float* b1 = (const float*)d_in[2];
    const float* W2 = (const float*)d_in[3];
    const float* b2 = (const float*)d_in[4];
    const int*   ei = (const int*)d_in[5];   // edge_index [2, E]

    const int C1 = in_sizes[2];              // 128
    const int C2 = in_sizes[4];              // 64
    const int N  = in_sizes[0] / C1;         // 100000
    const int E  = in_sizes[5] / 2;          // 1600000
    const int* src = ei;                     // edge_index[0] (sources)
    const int* dst = ei + E;                 // edge_index[1] (targets)

    // Workspace layout (floats): dinv[N] | h1[N*C1] | g1[N*C1] | h2[N*C2]
    float* ws   = (float*)d_ws;
    float* dinv = ws;
    float* h1   = dinv + (size_t)N;
    float* g1   = h1 + (size_t)N * C1;
    float* h2   = g1 + (size_t)N * C1;
    float* outp = (float*)d_out;

    // Symmetric normalization: deg from targets (+ self loops), dinv = rsqrt(deg).
    gcn_deg_init<<<(N + 255) / 256, 256, 0, stream>>>(dinv, N);
    gcn_deg_acc <<<(E + 255) / 256, 256, 0, stream>>>(dst, dinv, E);
    gcn_deg_fin <<<(N + 255) / 256, 256, 0, stream>>>(dinv, N);

    const int mblk = (N + 15) / 16;
    const int n4_1 = N * (C1 / 4);
    const int n4_2 = N * (C2 / 4);

    // Layer 1: h1 = x @ W1 ; g1 = self + b1 + edges ; ReLU.
    gcn_gemm_wmma<128><<<mblk, 128, 0, stream>>>(x, W1, h1, N);
    gcn_self_init<<<(n4_1 + 255) / 256, 256, 0, stream>>>(h1, b1, dinv, g1, n4_1, 5);
    gcn_edge_scatter<<<2048, 256, 0, stream>>>(h1, src, dst, dinv, g1, E, C1);
    gcn_relu<<<(n4_1 + 255) / 256, 256, 0, stream>>>(g1, n4_1);

    // Layer 2: h2 = g1 @ W2 ; d_out = self + b2 + edges.
    gcn_gemm_wmma<64><<<mblk, 128, 0, stream>>>(g1, W2, h2, N);
    gcn_self_init<<<(n4_2 + 255) / 256, 256, 0, stream>>>(h2, b2, dinv, outp, n4_2, 4);
    gcn_edge_scatter<<<2048, 256, 0, stream>>>(h2, src, dst, dinv, outp, E, C2);
}